// BiquadWQFilter_31628139167985
// MI455X (gfx1250) — compile-verified
//
#include <hip/hip_runtime.h>
#include <cmath>
#include <cstdint>

#ifndef M_PI
#define M_PI 3.14159265358979323846
#endif

// Problem shape from reference: (B, N) = (64, 131072), float32.
#define B_DIM     64
#define N_DIM     131072
#define CHUNK_L   128                      // samples per chunk-thread
#define C_PER_ROW (N_DIM / CHUNK_L)        // 1024 chunks per row
#define NTHREADS  (B_DIM * C_PER_ROW)      // 65536 chunk-threads
#define BLOCK1    64                       // 2 waves/block; LDS stage = 33.8 KB/block
#define SEG       (C_PER_ROW / 32)         // 32 chunks per lane in the pass-2 wave scan
#define XS_STRIDE 132                      // padded floats/thread: 528B, 16B-aligned, reduces bank conflicts

// ---- CDNA5 async global->LDS (device pass only; clean fallback otherwise) ----
#if defined(__HIP_DEVICE_COMPILE__)
#if __has_builtin(__builtin_amdgcn_global_load_async_to_lds_b128)
#define HAVE_ASYNC_LDS 1
#endif
#if __has_builtin(__builtin_amdgcn_s_wait_asynccnt)
#define HAVE_WAIT_ASYNC 1
#endif
#endif

#ifdef HAVE_ASYNC_LDS
// Builtin signature (from clang diagnostic / upstream "vV4i*1V4i*3IiIi"):
//   void __builtin_amdgcn_global_load_async_to_lds_b128(v4i as(1)* src, v4i as(3)* dst, imm off, imm cpol)
typedef int v4i_t __attribute__((vector_size(16)));
typedef __attribute__((address_space(1))) v4i_t gv4i_t;
typedef __attribute__((address_space(3))) v4i_t lv4i_t;
#endif

// RBJ lowpass coefficients, log-modulated w and Q. alpha = sin(w)*0.5*exp(-(logMinQ + q*QR))
// avoids the divide by q; single v_rcp for 1/a0. 5 transcendentals/sample total.
__device__ __forceinline__ void biquad_coef(float wm, float qm,
                                            float LMW, float WR, float NLMQ, float NQR,
                                            float& a1, float& a2, float& b0)
{
    wm = fminf(fmaxf(wm, 0.001f), 0.999f);
    qm = fminf(fmaxf(qm, 0.001f), 0.999f);
    float w     = __expf(fmaf(wm, WR, LMW));
    float inv2q = 0.5f * __expf(fmaf(qm, NQR, NLMQ));
    float s     = __sinf(w);
    float c     = __cosf(w);
    float alpha = s * inv2q;
    float ra0   = __builtin_amdgcn_rcpf(1.0f + alpha);
    a1 = -2.0f * c * ra0;
    a2 = (1.0f - alpha) * ra0;
    b0 = 0.5f * (1.0f - c) * ra0;
}

// ---------------- Pass 1: per-chunk affine aggregate (T 2x2, v 2) ----------------
// s' = M[n] s + [x,0],  M[n] = [[-a1,-a2],[1,0]]. Aggregates stored AoS: 6 floats/chunk.
__global__ __launch_bounds__(BLOCK1)
void biquad_pass1(const float* __restrict__ x, const float* __restrict__ wsig,
                  const float* __restrict__ qsig, float* __restrict__ agg,
                  float LMW, float WR, float NLMQ, float NQR)
{
#ifdef HAVE_ASYNC_LDS
    __shared__ float xs[BLOCK1 * XS_STRIDE];
#endif
    const int t = blockIdx.x * BLOCK1 + threadIdx.x;
    const int b = t / C_PER_ROW;
    const int c = t % C_PER_ROW;
    const size_t base = (size_t)b * N_DIM + (size_t)c * CHUNK_L;

    const float4* __restrict__ x4 = (const float4*)(x + base);
    const float4* __restrict__ w4 = (const float4*)(wsig + base);
    const float4* __restrict__ q4 = (const float4*)(qsig + base);

#ifdef HAVE_ASYNC_LDS
    {
        float* dst = &xs[threadIdx.x * XS_STRIDE];
        #pragma unroll
        for (int k = 0; k < CHUNK_L / 4; ++k) {
            __builtin_amdgcn_global_load_async_to_lds_b128(
                (gv4i_t*)(x4 + k), (lv4i_t*)(dst + 4 * k), 0, 0);
        }
    }
#endif

    float t00 = 1.f, t01 = 0.f, t10 = 0.f, t11 = 0.f, v0 = 0.f, v1 = 0.f;

#ifdef HAVE_ASYNC_LDS
#ifdef HAVE_WAIT_ASYNC
    __builtin_amdgcn_s_wait_asynccnt(0);
#else
    asm volatile("s_wait_asynccnt 0" ::: "memory");
#endif
    const float* xsrc = &xs[threadIdx.x * XS_STRIDE];
#endif

    for (int k = 0; k < CHUNK_L / 4; ++k) {
        float4 wv = w4[k];
        float4 qv = q4[k];
#ifdef HAVE_ASYNC_LDS
        float xv[4] = { xsrc[4*k+0], xsrc[4*k+1], xsrc[4*k+2], xsrc[4*k+3] };
#else
        float4 xq = x4[k];
        float xv[4] = { xq.x, xq.y, xq.z, xq.w };
#endif
        #pragma unroll
        for (int j = 0; j < 4; ++j) {
            float a1, a2, b0;
            biquad_coef(((const float*)&wv)[j], ((const float*)&qv)[j],
                        LMW, WR, NLMQ, NQR, a1, a2, b0);
            float u0 = fmaf(-a1, t00, -a2 * t10);
            float u1 = fmaf(-a1, t01, -a2 * t11);
            float uv = fmaf(-a1, v0, fmaf(-a2, v1, xv[j]));
            t10 = t00; t11 = t01; v1 = v0;
            t00 = u0;  t01 = u1;  v0 = uv;
        }
    }
    float* o = agg + (size_t)t * 6;
    o[0] = t00; o[1] = t01; o[2] = t10; o[3] = t11; o[4] = v0; o[5] = v1;
}

// ---------------- Pass 2: one wave32 per row, scan over chunk aggregates ----------------
// Initial filter state is zero, so each chunk's entry state = v-part of the exclusive prefix.
__global__ __launch_bounds__(32)
void biquad_pass2(const float* __restrict__ agg, float* __restrict__ ent)
{
    const int b = blockIdx.x;
    const int lane = threadIdx.x;
    const float* __restrict__ ag = agg + (size_t)b * C_PER_ROW * 6;
    const int c0 = lane * SEG;

    // serial compose of this lane's segment (time order)
    float T00 = 1.f, T01 = 0.f, T10 = 0.f, T11 = 0.f, V0 = 0.f, V1 = 0.f;
    for (int j = 0; j < SEG; ++j) {
        const float* a = ag + (size_t)(c0 + j) * 6;
        float a00 = a[0], a01 = a[1], a10 = a[2], a11 = a[3], av0 = a[4], av1 = a[5];
        float n00 = fmaf(a00, T00, a01 * T10);
        float n01 = fmaf(a00, T01, a01 * T11);
        float n10 = fmaf(a10, T00, a11 * T10);
        float n11 = fmaf(a10, T01, a11 * T11);
        float nv0 = fmaf(a00, V0, fmaf(a01, V1, av0));
        float nv1 = fmaf(a10, V0, fmaf(a11, V1, av1));
        T00 = n00; T01 = n01; T10 = n10; T11 = n11; V0 = nv0; V1 = nv1;
    }

    // Kogge-Stone inclusive scan across 32 lanes (lower lanes = earlier time, applied first)
    for (int d = 1; d < 32; d <<= 1) {
        float o00 = __shfl_up(T00, d), o01 = __shfl_up(T01, d);
        float o10 = __shfl_up(T10, d), o11 = __shfl_up(T11, d);
        float ov0 = __shfl_up(V0, d),  ov1 = __shfl_up(V1, d);
        if (lane >= d) {
            float n00 = fmaf(T00, o00, T01 * o10);
            float n01 = fmaf(T00, o01, T01 * o11);
            float n10 = fmaf(T10, o00, T11 * o10);
            float n11 = fmaf(T10, o01, T11 * o11);
            float nv0 = fmaf(T00, ov0, fmaf(T01, ov1, V0));
            float nv1 = fmaf(T10, ov0, fmaf(T11, ov1, V1));
            T00 = n00; T01 = n01; T10 = n10; T11 = n11; V0 = nv0; V1 = nv1;
        }
    }

    // exclusive prefix applied to zero initial state = its v-part
    float s0 = __shfl_up(V0, 1);
    float s1 = __shfl_up(V1, 1);
    if (lane == 0) { s0 = 0.f; s1 = 0.f; }

    // walk the segment emitting per-chunk entry states
    float* __restrict__ en = ent + (size_t)b * C_PER_ROW * 2;
    for (int j = 0; j < SEG; ++j) {
        const int ci = c0 + j;
        en[ci * 2 + 0] = s0;
        en[ci * 2 + 1] = s1;
        const float* a = ag + (size_t)ci * 6;
        float ns0 = fmaf(a[0], s0, fmaf(a[1], s1, a[4]));
        float ns1 = fmaf(a[2], s0, fmaf(a[3], s1, a[5]));
        s0 = ns0; s1 = ns1;
    }
}

// ---------------- Pass 3: re-run recursion from entry state, apply FIR, write out ----------------
// Inputs are L2-resident after pass 1 (133 MB working set < 192 MB L2), so the re-read is cheap.
// FIR uses b1 = 2*b0, b2 = b0:  out = b0*(y + 2*y[n-1] + y[n-2]).
__global__ __launch_bounds__(BLOCK1)
void biquad_pass3(const float* __restrict__ x, const float* __restrict__ wsig,
                  const float* __restrict__ qsig, const float* __restrict__ ent,
                  float* __restrict__ out,
                  float LMW, float WR, float NLMQ, float NQR)
{
#ifdef HAVE_ASYNC_LDS
    __shared__ float xs[BLOCK1 * XS_STRIDE];
#endif
    const int t = blockIdx.x * BLOCK1 + threadIdx.x;
    const int b = t / C_PER_ROW;
    const int c = t % C_PER_ROW;
    const size_t base = (size_t)b * N_DIM + (size_t)c * CHUNK_L;

    const float4* __restrict__ x4 = (const float4*)(x + base);
    const float4* __restrict__ w4 = (const float4*)(wsig + base);
    const float4* __restrict__ q4 = (const float4*)(qsig + base);
    float4* __restrict__ o4 = (float4*)(out + base);

#ifdef HAVE_ASYNC_LDS
    {
        float* dst = &xs[threadIdx.x * XS_STRIDE];
        #pragma unroll
        for (int k = 0; k < CHUNK_L / 4; ++k) {
            __builtin_amdgcn_global_load_async_to_lds_b128(
                (gv4i_t*)(x4 + k), (lv4i_t*)(dst + 4 * k), 0, 0);
        }
    }
#endif

    float y1 = ent[(size_t)t * 2 + 0];   // y[n-1] entering this chunk
    float y2 = ent[(size_t)t * 2 + 1];   // y[n-2] entering this chunk

#ifdef HAVE_ASYNC_LDS
#ifdef HAVE_WAIT_ASYNC
    __builtin_amdgcn_s_wait_asynccnt(0);
#else
    asm volatile("s_wait_asynccnt 0" ::: "memory");
#endif
    const float* xsrc = &xs[threadIdx.x * XS_STRIDE];
#endif

    for (int k = 0; k < CHUNK_L / 4; ++k) {
        float4 wv = w4[k];
        float4 qv = q4[k];
#ifdef HAVE_ASYNC_LDS
        float xv[4] = { xsrc[4*k+0], xsrc[4*k+1], xsrc[4*k+2], xsrc[4*k+3] };
#else
        float4 xq = x4[k];
        float xv[4] = { xq.x, xq.y, xq.z, xq.w };
#endif
        float4 ov;
        #pragma unroll
        for (int j = 0; j < 4; ++j) {
            float a1, a2, b0;
            biquad_coef(((const float*)&wv)[j], ((const float*)&qv)[j],
                        LMW, WR, NLMQ, NQR, a1, a2, b0);
            float y = fmaf(-a1, y1, fmaf(-a2, y2, xv[j]));
            ((float*)&ov)[j] = b0 * fmaf(2.0f, y1, y + y2);
            y2 = y1; y1 = y;
        }
        o4[k] = ov;
    }
}

extern "C" void kernel_launch(void* const* d_in, const int* in_sizes, int n_in,
                              void* d_out, int out_size, void* d_ws, size_t ws_size,
                              hipStream_t stream) {
    (void)in_sizes; (void)n_in; (void)out_size; (void)ws_size;
    const float* x = (const float*)d_in[0];
    const float* w = (const float*)d_in[1];
    const float* q = (const float*)d_in[2];
    float* out = (float*)d_out;

    // Workspace: 6 floats/chunk aggregates + 2 floats/chunk entry states = 2 MB total.
    float* agg = (float*)d_ws;
    float* ent = agg + (size_t)NTHREADS * 6;

    // Log-domain modulation constants in double precision on host.
    const double minw = 2.0 * M_PI * 20.0 / 44100.0;
    const double maxw = 2.0 * M_PI * 8000.0 / 44100.0;
    const float LMW  = (float)std::log(minw);
    const float WR   = (float)(std::log(maxw) - std::log(minw));
    const float NLMQ = (float)(-std::log(0.7071));
    const float NQR  = (float)(-(std::log(8.0) - std::log(0.7071)));

    biquad_pass1<<<NTHREADS / BLOCK1, BLOCK1, 0, stream>>>(x, w, q, agg, LMW, WR, NLMQ, NQR);
    biquad_pass2<<<B_DIM, 32, 0, stream>>>(agg, ent);
    biquad_pass3<<<NTHREADS / BLOCK1, BLOCK1, 0, stream>>>(x, w, q, ent, out, LMW, WR, NLMQ, NQR);
}